// TTN_VQC_90202903150770
// MI455X (gfx1250) — compile-verified
//
#include <hip/hip_runtime.h>

// ---------------------------------------------------------------------------
// TTN_VQC on MI455X (gfx1250, wave32)
//   16-qubit statevector sim, batch 16, depth 6, + linear head via WMMA.
//
// Strategy (L2-resident, pass-fused):
//   1) encode_kernel      : closed-form product state (1 write pass)
//   per layer l = 0..5:
//   2) gates_lo_kernel    : CNOT-ring permutation fused into gather +
//                           rotations on qubits 0..10 inside a 16KB LDS tile
//   3) gates_hi_kernel    : rotations on qubits 11..15, in-place, re-tiled;
//                           tile moved by the Tensor Data Mover (TDM)
//   4) expect_kernel      : <Z_q> per sample (signed |amp|^2 reduction)
//   5) head_wmma_kernel   : out = exps @ W^T + b via v_wmma_f32_16x16x4_f32
// ---------------------------------------------------------------------------

#define NQ      16
#define DEPTH   6
#define BATCH   16
#define FEAT    10
#define NSTATES 65536        // 2^16 amplitudes per sample

typedef __attribute__((ext_vector_type(2))) float        v2f;
typedef __attribute__((ext_vector_type(8))) float        v8f;
typedef __attribute__((ext_vector_type(4))) unsigned int u32x4;
typedef __attribute__((ext_vector_type(4))) int          i32x4;
typedef __attribute__((ext_vector_type(8))) int          i32x8;

#if defined(__has_builtin)
#if __has_builtin(__builtin_amdgcn_tensor_load_to_lds) && \
    __has_builtin(__builtin_amdgcn_tensor_store_from_lds) && \
    __has_builtin(__builtin_amdgcn_s_wait_tensorcnt)
#define HAVE_TDM 1
#endif
#endif

// Inverse of the CNOT-ring basis permutation sigma.
// Forward: y_q = x_0 ^ ... ^ x_q (q=1..15), y_0 = x_1 ^ ... ^ x_15.
// Inverse: x_q = y_q ^ y_{q-1} (q=2..15), x_1 = y_0^y_1^y_15, x_0 = y_0^y_15.
__device__ __forceinline__ unsigned inv_sigma(unsigned y) {
    unsigned y15 = (y >> 15) & 1u;
    unsigned t   = (y ^ (y << 1)) & 0xFFFCu;           // bits 2..15
    unsigned b0  = (y & 1u) ^ y15;                     // bit 0
    unsigned b1  = (((y >> 1) ^ y) & 1u) ^ y15;        // bit 1
    return t | b0 | (b1 << 1);
}

// Fused single-qubit gate G = Rz(2hz) * Ry(2hy) * Rx(2hx) (half-angles in).
__device__ __forceinline__ void fused_gate(float hx, float hy, float hz,
                                           float2& g00, float2& g01,
                                           float2& g10, float2& g11) {
    float cx = __cosf(hx), sx = __sinf(hx);
    float cy = __cosf(hy), sy = __sinf(hy);
    float cz = __cosf(hz), sz = __sinf(hz);
    float cycx = cy * cx, sysx = sy * sx;
    float sycx = sy * cx, cysx = cy * sx;
    g00 = make_float2( cz * cycx + sz * sysx,  cz * sysx - sz * cycx);
    g01 = make_float2(-cz * sycx - sz * cysx, -cz * cysx + sz * sycx);
    g10 = make_float2( cz * sycx + sz * cysx, -cz * cysx + sz * sycx);
    g11 = make_float2( cz * cycx + sz * sysx,  sz * cycx - cz * sysx);
}

#ifdef HAVE_TDM
// Build a 2D tile D# and issue a TDM load/store between global and LDS.
//   tile = tile_d1 rows x tile_d0 elements of 8B (float2), row stride
//   `stride_elems` elements in memory, densely packed in LDS at `ldsoff`.
// D# bit layout per CDNA5 ISA 8.3/8.4 (group0: count/lds/global/type,
// group1: data_size, dims, tile dims, strides). Groups 2/3 unused (2D).
__device__ __forceinline__ void tdm_tile_2d(const void* gaddr, unsigned ldsoff,
                                            int tile_d0, int tile_d1,
                                            int stride_elems, bool store) {
    unsigned long long ga = (unsigned long long)gaddr;
    u32x4 g0;
    g0.x = 1u;                                        // count=1, user descriptor
    g0.y = ldsoff;                                    // LDS byte address
    g0.z = (unsigned)(ga & 0xFFFFFFFFull);            // global_addr[31:0]
    g0.w = (unsigned)((ga >> 32) & 0x1FFFFFFull)      // global_addr[56:32]
         | (2u << 30);                                // type = 2 ("image")
    i32x8 g1;
    g1[0] = 3 << 16;                                  // data_size=3 (8B), mask=0
    g1[1] = tile_d0 << 16;                            // tensor_dim0 [79:48]
    g1[2] = tile_d1 << 16;                            // tensor_dim1 [111:80]
    g1[3] = tile_d0 << 16;                            // tile_dim0   [127:112]
    g1[4] = tile_d1;                                  // tile_dim1   [143:128]
    g1[5] = stride_elems;                             // tensor_dim0_stride low
    g1[6] = 0;
    g1[7] = 0;
    i32x4 z4 = {};
#if __clang_major__ >= 23
    i32x8 z8 = {};
    if (store)
        __builtin_amdgcn_tensor_store_from_lds(g0, g1, z4, z4, z8, 0);
    else
        __builtin_amdgcn_tensor_load_to_lds(g0, g1, z4, z4, z8, 0);
#else
    if (store)
        __builtin_amdgcn_tensor_store_from_lds(g0, g1, z4, z4, 0);
    else
        __builtin_amdgcn_tensor_load_to_lds(g0, g1, z4, z4, 0);
#endif
}
#endif

// --------------------------------------------------------------------------
// 1) Encoding layer: product state, amp(x) = prod_q (x_q ? sin : cos)
// --------------------------------------------------------------------------
__global__ void encode_kernel(const float* __restrict__ feat,
                              float2* __restrict__ S) {
    const int b   = blockIdx.x;
    const int tid = threadIdx.x;                 // 1024 threads
    __shared__ float cs[NQ], sn[NQ];
    if (tid < NQ) {
        // q_in = feat * pi/2 ; RY half-angle = q_in/2 = feat * pi/4
        float h = feat[b * NQ + tid] * 0.78539816339744830962f;
        cs[tid] = __cosf(h);
        sn[tid] = __sinf(h);
    }
    __syncthreads();
    const size_t base = ((size_t)b) << 16;
    for (int k = 0; k < NSTATES / 1024; ++k) {
        unsigned idx = (unsigned)(k * 1024 + tid);
        float a = 1.0f;
#pragma unroll
        for (int q = 0; q < NQ; ++q)
            a *= ((idx >> q) & 1u) ? sn[q] : cs[q];
        S[base + idx] = make_float2(a, 0.0f);
    }
}

// --------------------------------------------------------------------------
// 2) Low-qubit pass: permuted gather (CNOT ring) + rotations on qubits 0..10
//    grid = BATCH*32 blocks (bits 11..15 fixed per block), 256 threads
// --------------------------------------------------------------------------
#define LO_Q    11
#define TILE    2048
__global__ void gates_lo_kernel(const float2* __restrict__ Sin,
                                float2* __restrict__ Sout,
                                const float* __restrict__ qp, int layer) {
    __shared__ float2 tile[TILE];
    __shared__ float2 G[LO_Q][4];
    const int      b   = blockIdx.x >> 5;
    const unsigned yhi = (unsigned)(blockIdx.x & 31) << 11;
    const int      tid = threadIdx.x;            // 256

    if (tid < LO_Q) {
        int q = tid;
        float hx = 0.5f * qp[  0 + layer * 16 + q];
        float hy = 0.5f * qp[ 96 + layer * 16 + q];
        float hz = 0.5f * qp[192 + layer * 16 + q];
        fused_gate(hx, hy, hz, G[q][0], G[q][1], G[q][2], G[q][3]);
    }

    const size_t base = ((size_t)b) << 16;
#pragma unroll
    for (int k = 0; k < 8; ++k) {
        unsigned ylo = (unsigned)(k * 256 + tid);
        unsigned x   = inv_sigma(yhi | ylo);     // fused CNOT-ring permutation
        tile[ylo]    = Sin[base + x];
    }
    __syncthreads();

    for (int q = 0; q < LO_Q; ++q) {
        float2 g00 = G[q][0], g01 = G[q][1], g10 = G[q][2], g11 = G[q][3];
#pragma unroll
        for (int k = 0; k < 4; ++k) {
            unsigned p  = (unsigned)(k * 256 + tid);        // 1024 pairs
            unsigned i0 = ((p >> q) << (q + 1)) | (p & ((1u << q) - 1u));
            unsigned i1 = i0 | (1u << q);
            float2 a0 = tile[i0], a1 = tile[i1];
            float2 n0, n1;
            n0.x = g00.x * a0.x - g00.y * a0.y + g01.x * a1.x - g01.y * a1.y;
            n0.y = g00.x * a0.y + g00.y * a0.x + g01.x * a1.y + g01.y * a1.x;
            n1.x = g10.x * a0.x - g10.y * a0.y + g11.x * a1.x - g11.y * a1.y;
            n1.y = g10.x * a0.y + g10.y * a0.x + g11.x * a1.y + g11.y * a1.x;
            tile[i0] = n0;
            tile[i1] = n1;
        }
        __syncthreads();
    }

    // Contiguous write-back (dense 2048-element run)
#pragma unroll
    for (int k = 0; k < 8; ++k) {
        unsigned ylo = (unsigned)(k * 256 + tid);
        Sout[base + (yhi | ylo)] = tile[ylo];
    }
}

// --------------------------------------------------------------------------
// 3) High-qubit pass: rotations on qubits 11..15, in-place.
//    Tile-local bits: [0..5] = amp bits 0..5, [6..10] = amp bits 11..15.
//    Tile in memory = 32 rows x 64 float2, row stride 2048 elements -> TDM.
//    grid = BATCH*32 blocks (bits 6..10 fixed per block), 256 threads
// --------------------------------------------------------------------------
__global__ void gates_hi_kernel(float2* __restrict__ S,
                                const float* __restrict__ qp, int layer) {
    __shared__ float2 tile[TILE];
    __shared__ float2 G[5][4];
    const int      b   = blockIdx.x >> 5;
    const unsigned mid = (unsigned)(blockIdx.x & 31) << 6;   // bits 6..10
    const int      tid = threadIdx.x;            // 256

    if (tid < 5) {
        int q = 11 + tid;
        float hx = 0.5f * qp[  0 + layer * 16 + q];
        float hy = 0.5f * qp[ 96 + layer * 16 + q];
        float hz = 0.5f * qp[192 + layer * 16 + q];
        fused_gate(hx, hy, hz, G[tid][0], G[tid][1], G[tid][2], G[tid][3]);
    }

    const size_t base = ((size_t)b) << 16;
    float2* gbase = S + base + mid;              // tile start in global memory

#ifdef HAVE_TDM
    // Wave 0 drives the TDM: one descriptor moves the whole 16KB tile.
    if (tid < 32) {
        tdm_tile_2d(gbase, (unsigned)(size_t)&tile[0],
                    /*tile_d0=*/64, /*tile_d1=*/32,
                    /*stride_elems=*/2048, /*store=*/false);
        __builtin_amdgcn_s_wait_tensorcnt(0);
    }
    __syncthreads();
#else
#pragma unroll
    for (int k = 0; k < 8; ++k) {
        unsigned j = (unsigned)(k * 256 + tid);
        tile[j] = gbase[((j >> 6) << 11) | (j & 63u)];
    }
    __syncthreads();
#endif

    for (int t = 0; t < 5; ++t) {
        const int ql = 6 + t;                    // tile-local bit for qubit 11+t
        float2 g00 = G[t][0], g01 = G[t][1], g10 = G[t][2], g11 = G[t][3];
#pragma unroll
        for (int k = 0; k < 4; ++k) {
            unsigned p  = (unsigned)(k * 256 + tid);
            unsigned i0 = ((p >> ql) << (ql + 1)) | (p & ((1u << ql) - 1u));
            unsigned i1 = i0 | (1u << ql);
            float2 a0 = tile[i0], a1 = tile[i1];
            float2 n0, n1;
            n0.x = g00.x * a0.x - g00.y * a0.y + g01.x * a1.x - g01.y * a1.y;
            n0.y = g00.x * a0.y + g00.y * a0.x + g01.x * a1.y + g01.y * a1.x;
            n1.x = g10.x * a0.x - g10.y * a0.y + g11.x * a1.x - g11.y * a1.y;
            n1.y = g10.x * a0.y + g10.y * a0.x + g11.x * a1.y + g11.y * a1.x;
            tile[i0] = n0;
            tile[i1] = n1;
        }
        __syncthreads();
    }

#ifdef HAVE_TDM
    // LDS is consistent after the last barrier; wave 0 streams the tile out.
    if (tid < 32) {
        tdm_tile_2d(gbase, (unsigned)(size_t)&tile[0],
                    64, 32, 2048, /*store=*/true);
        __builtin_amdgcn_s_wait_tensorcnt(0);
    }
#else
#pragma unroll
    for (int k = 0; k < 8; ++k) {
        unsigned j = (unsigned)(k * 256 + tid);
        gbase[((j >> 6) << 11) | (j & 63u)] = tile[j];
    }
#endif
}

// --------------------------------------------------------------------------
// 4) Expectation values: exps[b][q] = sum_x sign(bit q of x) * |amp|^2
// --------------------------------------------------------------------------
__global__ void expect_kernel(const float2* __restrict__ S,
                              float* __restrict__ exps) {
    const int b   = blockIdx.x;
    const int tid = threadIdx.x;                 // 256
    const size_t base = ((size_t)b) << 16;
    float acc[NQ];
#pragma unroll
    for (int q = 0; q < NQ; ++q) acc[q] = 0.0f;

    for (int k = 0; k < NSTATES / 256; ++k) {
        unsigned idx = (unsigned)(k * 256 + tid);
        float2 a = S[base + idx];
        float  p = a.x * a.x + a.y * a.y;
#pragma unroll
        for (int q = 0; q < NQ; ++q)
            acc[q] += ((idx >> q) & 1u) ? -p : p;
    }

    __shared__ float red[256];
    for (int q = 0; q < NQ; ++q) {
        red[tid] = acc[q];
        __syncthreads();
        for (int s = 128; s > 0; s >>= 1) {
            if (tid < s) red[tid] += red[tid + s];
            __syncthreads();
        }
        if (tid == 0) exps[b * NQ + q] = red[0];
        __syncthreads();
    }
}

// --------------------------------------------------------------------------
// 5) Linear head via WMMA: out(16x10) = exps(16x16) @ W^T(16x10) + b
//    One wave32, four V_WMMA_F32_16X16X4_F32 accumulations over K=16.
//    A 16x4 layout: lanes 0-15 hold M=0-15; VGPR0 = K {0 | 2}, VGPR1 = K {1 | 3}
//    B 4x16 layout: lanes hold N; VGPR0 = K row {0 | 2}, VGPR1 = K row {1 | 3}
//    C/D: VGPR r -> (M=r, N=lane) for lanes 0-15, (M=r+8, N=lane-16) else.
//    B-padding (N >= 10 -> 0) via clamped in-bounds load + multiply mask,
//    so no EXEC-mask branching around the loads.
// --------------------------------------------------------------------------
__global__ void head_wmma_kernel(const float* __restrict__ exps,
                                 const float* __restrict__ W,
                                 const float* __restrict__ bias,
                                 float* __restrict__ out) {
    const int  lane   = threadIdx.x;             // 32 threads
    const bool hiHalf = lane >= 16;
    const int  l16    = lane & 15;

    const int   nclamp = (l16 < FEAT) ? l16 : (FEAT - 1);
    const float nmask  = (l16 < FEAT) ? 1.0f : 0.0f;

    v8f acc = {};
#pragma unroll
    for (int k0 = 0; k0 < 16; k0 += 4) {
        const int ka = k0 + (hiHalf ? 2 : 0);
        v2f A, B;
        A.x = exps[l16 * 16 + ka];               // A[m][k]   = exps[m][k]
        A.y = exps[l16 * 16 + ka + 1];
        B.x = nmask * W[nclamp * 16 + ka];       // B[k][n] = W[n][k], padded
        B.y = nmask * W[nclamp * 16 + ka + 1];
        acc = __builtin_amdgcn_wmma_f32_16x16x4_f32(
            /*neg_a=*/false, A, /*neg_b=*/false, B,
            /*c_mod=*/(short)0, acc, /*reuse_a=*/false, /*reuse_b=*/false);
    }

#pragma unroll
    for (int r = 0; r < 8; ++r) {
        int m = hiHalf ? (r + 8) : r;
        int n = l16;
        if (n < FEAT) out[m * FEAT + n] = acc[r] + bias[n];
    }
}

// --------------------------------------------------------------------------
extern "C" void kernel_launch(void* const* d_in, const int* in_sizes, int n_in,
                              void* d_out, int out_size, void* d_ws,
                              size_t ws_size, hipStream_t stream) {
    (void)in_sizes; (void)n_in; (void)out_size; (void)ws_size;

    const float* feat = (const float*)d_in[0];   // (16,16) f32
    const float* qp   = (const float*)d_in[1];   // (288,)  f32
    const float* W    = (const float*)d_in[2];   // (10,16) f32
    const float* bias = (const float*)d_in[3];   // (10,)   f32
    float*       out  = (float*)d_out;           // (16,10) f32

    // Workspace: S0 (8MB) | S1 (8MB) | exps (1KB)   -> ~16 MB of d_ws
    float2* S0   = (float2*)d_ws;
    float2* S1   = S0 + (size_t)BATCH * NSTATES;
    float*  exps = (float*)(S1 + (size_t)BATCH * NSTATES);

    encode_kernel<<<BATCH, 1024, 0, stream>>>(feat, S0);

    float2* a = S0;
    float2* b = S1;
    for (int l = 0; l < DEPTH; ++l) {
        gates_lo_kernel<<<BATCH * 32, 256, 0, stream>>>(a, b, qp, l);
        gates_hi_kernel<<<BATCH * 32, 256, 0, stream>>>(b, qp, l);
        float2* t = a; a = b; b = t;
    }

    expect_kernel<<<BATCH, 256, 0, stream>>>(a, exps);
    head_wmma_kernel<<<1, 32, 0, stream>>>(exps, W, bias, out);
}